// H2GCN_68143951118628
// MI455X (gfx1250) — compile-verified
//
#include <hip/hip_runtime.h>
#include <hip/hip_bf16.h>

// H2GCN forward for MI455X (gfx1250, wave32, WMMA).
// fp32 end-to-end; dense GEMMs use V_WMMA_F32_16X16X4_F32 with LDS-staged tiles.

typedef __attribute__((ext_vector_type(2))) float v2f;
typedef __attribute__((ext_vector_type(8))) float v8f;

#define NN 50000
#define EE 400000
#define DD 256
#define HH 256
#define CC 64

#define LDK  32    // K-chunk staged in LDS per block iteration
#define PRS  160   // sB pair-row stride in floats: 160 mod 64 == 32 -> half-waves
                   // use disjoint 32-bank sets for b64 fragment loads
#define ASTR 36    // sA row stride: 36*d mod 64 never hits {62,63,0,1,2,3} for d!=0

// ---------------------------------------------------------------- degree / dis
__global__ void deg_init_kernel(float* __restrict__ deg) {
    int i = blockIdx.x * blockDim.x + threadIdx.x;
    if (i < NN) deg[i] = 1.0f;   // self-loop contributes 1 to every node's degree
}

__global__ void deg_accum_kernel(const int* __restrict__ row, float* __restrict__ deg) {
    int e = blockIdx.x * blockDim.x + threadIdx.x;
    if (e < EE) atomicAdd(&deg[row[e]], 1.0f);
}

__global__ void dis_kernel(const float* __restrict__ deg, float* __restrict__ dis) {
    int i = blockIdx.x * blockDim.x + threadIdx.x;
    if (i < NN) {
        float d = deg[i];
        dis[i] = (d > 0.0f) ? rsqrtf(d) : 0.0f;   // deg >= 1 always (self loop)
    }
}

// -------------------------------------------------------------------- SpMM
// Self-loop term: agg[i][:] = dis[i]^2 * x[i][:]
__global__ void spmm_self_kernel(const float* __restrict__ x, const float* __restrict__ dis,
                                 float* __restrict__ agg) {
    int t = blockIdx.x * blockDim.x + threadIdx.x;      // over NN*DD (12.8M < 2^31)
    if (t < NN * DD) {
        int i = t / DD;
        float w = dis[i];
        agg[t] = (w * w) * x[t];
    }
}

// Edge scatter: agg[row][:] += dis[row]*dis[col] * x[col][:]
// One thread handles one edge x 4 features (float4 gather, 4 atomic f32 scatters).
__global__ void spmm_edge_kernel(const int* __restrict__ row, const int* __restrict__ col,
                                 const float* __restrict__ dis, const float* __restrict__ x,
                                 float* __restrict__ agg) {
    int t = blockIdx.x * blockDim.x + threadIdx.x;      // over EE*(DD/4) = 25.6M
    const int F4 = DD / 4;
    if (t >= EE * F4) return;
    int e = t / F4;
    int f = (t - e * F4) * 4;
    int r = row[e];
    int c = col[e];
    float w = dis[r] * dis[c];
    const float4 xs = *(const float4*)(x + (long long)c * DD + f);
    float* dst = agg + (long long)r * DD + f;
    atomicAdd(dst + 0, w * xs.x);
    atomicAdd(dst + 1, w * xs.y);
    atomicAdd(dst + 2, w * xs.z);
    atomicAdd(dst + 3, w * xs.w);
}

// -------------------------------------------------------------------- GEMM
// Y[:, col_off ...] = act(X[N x kdim](ldx) @ W[kdim x hout] + bias)
// Block = 128 threads = 4 waves; all share one 64-column W group per 32-K chunk in
// LDS; each wave owns TWO 16-row A tiles -> every B fragment feeds 2 WMMAs.
// sB is stored fragment-major: element (k,c) of the chunk lives at
//   sB[(k>>1)*PRS + 2*c + (k&1)]
// so a lane's B fragment {B[k][c], B[k+1][c]} is one aligned ds_load_b64 directly
// into the even-aligned VGPR pair WMMA wants (no repack movs); tiles t,t+1 pair
// into ds_load_2addr_b64.
//
// V_WMMA_F32_16X16X4_F32 VGPR layout (ISA 7.12.2):
//   A (16x4): lanes 0-15 hold {K,K+1} in v[0],v[1]; lanes 16-31 hold {K+2,K+3}.
//   B (4x16): row-striped: v[0]/v[1] lanes 0-15 -> rows K/K+1, lanes 16-31 -> K+2/K+3.
//   C/D (16x16): v[i] lane<16 -> M=i, N=lane; lane>=16 -> M=8+i, N=lane-16.
template <bool RELU>
__global__ __launch_bounds__(128)
void gemm_wmma_kernel(const float* __restrict__ X, int ldx,
                      const float* __restrict__ W,       // row-major [kdim x hout]
                      const float* __restrict__ bias,    // [hout]
                      float* __restrict__ Y, int ldy, int col_off,
                      int kdim, int hout, int ntiles) {
    __shared__ float sB[16 * PRS];           // 16 pair-rows x 64 cols interleaved, 10240 B
    __shared__ float sA[4][32 * ASTR];       // per-wave 2 tiles x 16 rows x 32 k, 18432 B

    const int tid  = threadIdx.x;
    const int lane = tid & 31;
    const int wave = tid >> 5;
    const int half = lane >> 4;              // 0: lanes 0-15, 1: lanes 16-31
    const int r    = lane & 15;

    int tb    = blockIdx.x * 8 + wave * 2;
    int tile0 = tb     < ntiles ? tb     : ntiles - 1;   // clamp: idempotent dup writes
    int tile1 = tb + 1 < ntiles ? tb + 1 : ntiles - 1;
    const int row0 = tile0 * 16;
    const int row1 = tile1 * 16;
    const int col0 = blockIdx.y * 64;        // 4 column sub-tiles of 16

    float* sAw = sA[wave];

    v8f acc[2][4];
#pragma unroll
    for (int i = 0; i < 2; ++i)
#pragma unroll
        for (int t = 0; t < 4; ++t) acc[i][t] = v8f{};

    for (int k0 = 0; k0 < kdim; k0 += LDK) {
        // ---- stage B chunk (fragment-major): 16 pair-rows x 16 col4-groups = 256 tasks
#pragma unroll
        for (int task = tid; task < 256; task += 128) {
            const int p  = task >> 4;          // pair-row 0..15  (k rows 2p, 2p+1)
            const int c4 = (task & 15) * 4;    // column group within 64
            const float* w0 = W + (size_t)(k0 + 2 * p) * hout + col0 + c4;
            float4 wa = *(const float4*)(w0);
            float4 wb = *(const float4*)(w0 + hout);
            float4 lo = make_float4(wa.x, wb.x, wa.y, wb.y);
            float4 hi = make_float4(wa.z, wb.z, wa.w, wb.w);
            float* d = &sB[p * PRS + c4 * 2];
            *(float4*)(d + 0) = lo;            // ds_store_b128
            *(float4*)(d + 4) = hi;
        }
        // ---- stage A tiles: 2 x 16 rows x 32 k, coalesced 128B per row
#pragma unroll 4
        for (int rr = 0; rr < 16; ++rr) {
            sAw[rr * ASTR + lane]        = X[(size_t)(row0 + rr) * ldx + k0 + lane];
            sAw[(16 + rr) * ASTR + lane] = X[(size_t)(row1 + rr) * ldx + k0 + lane];
        }
        __syncthreads();

        // ---- 8 k-steps x 8 WMMA from LDS
#pragma unroll
        for (int kk = 0; kk < LDK; kk += 4) {
            v2f a0 = *(const v2f*)&sAw[r * ASTR + kk + 2 * half];          // ds_load_b64
            v2f a1 = *(const v2f*)&sAw[(16 + r) * ASTR + kk + 2 * half];
            const int pb = ((kk >> 1) + half) * PRS;                       // pair-row base
#pragma unroll
            for (int t = 0; t < 4; ++t) {
                v2f b = *(const v2f*)&sB[pb + (t * 16 + r) * 2];           // ds_load_b64
                acc[0][t] = __builtin_amdgcn_wmma_f32_16x16x4_f32(
                    false, a0, false, b, (short)0, acc[0][t], false, false);
                acc[1][t] = __builtin_amdgcn_wmma_f32_16x16x4_f32(
                    false, a1, false, b, (short)0, acc[1][t], false, false);
            }
        }
        __syncthreads();
    }

    // ---- epilogue: bias (+ ReLU), scatter per C/D layout, both row tiles
#pragma unroll
    for (int t = 0; t < 4; ++t) {
        int c    = col0 + t * 16 + r;
        float bv = bias[c];
#pragma unroll
        for (int i = 0; i < 8; ++i) {
            int m0 = row0 + i + 8 * half;
            int m1 = row1 + i + 8 * half;
            float v0 = acc[0][t][i] + bv;
            float v1 = acc[1][t][i] + bv;
            if (RELU) { v0 = v0 > 0.0f ? v0 : 0.0f; v1 = v1 > 0.0f ? v1 : 0.0f; }
            Y[(size_t)m0 * ldy + col_off + c] = v0;
            Y[(size_t)m1 * ldy + col_off + c] = v1;
        }
    }
}

// -------------------------------------------------------------------- launch
extern "C" void kernel_launch(void* const* d_in, const int* in_sizes, int n_in,
                              void* d_out, int out_size, void* d_ws, size_t ws_size,
                              hipStream_t stream) {
    (void)in_sizes; (void)n_in; (void)out_size; (void)ws_size;
    const float* x  = (const float*)d_in[0];
    const int*   ei = (const int*)d_in[1];   // [2,E] row-major: rows then cols
    // d_in[2] = num_nodes scalar (compile-time NN used)
    const float* W0 = (const float*)d_in[3];
    const float* b0 = (const float*)d_in[4];
    const float* W1 = (const float*)d_in[5];
    const float* b1 = (const float*)d_in[6];
    const float* W2 = (const float*)d_in[7];
    const float* b2 = (const float*)d_in[8];
    const float* Wc = (const float*)d_in[9];
    const float* bc = (const float*)d_in[10];
    float* out = (float*)d_out;

    // workspace: deg[N] | dis[N] | agg1[N*D] | agg2[N*D] | hcat[N*3H]  (~257 MB)
    float* deg  = (float*)d_ws;
    float* dis  = deg + NN;
    float* agg1 = dis + NN;
    float* agg2 = agg1 + (size_t)NN * DD;
    float* hcat = agg2 + (size_t)NN * DD;

    const int* erow = ei;
    const int* ecol = ei + EE;

    // normalized adjacency weights
    deg_init_kernel<<<(NN + 255) / 256, 256, 0, stream>>>(deg);
    deg_accum_kernel<<<(EE + 255) / 256, 256, 0, stream>>>(erow, deg);
    dis_kernel<<<(NN + 255) / 256, 256, 0, stream>>>(deg, dis);

    // hop 1: agg1 = A_norm @ x
    spmm_self_kernel<<<(NN * DD) / 256, 256, 0, stream>>>(x, dis, agg1);
    spmm_edge_kernel<<<(EE * (DD / 4) + 255) / 256, 256, 0, stream>>>(erow, ecol, dis, x, agg1);
    // hop 2: agg2 = A_norm @ agg1
    spmm_self_kernel<<<(NN * DD) / 256, 256, 0, stream>>>(agg1, dis, agg2);
    spmm_edge_kernel<<<(EE * (DD / 4) + 255) / 256, 256, 0, stream>>>(erow, ecol, dis, agg1, agg2);

    const int ntiles = NN / 16;                    // 3125 row tiles
    const int gx     = (ntiles + 7) / 8;           // 4 waves x 2 tiles per block

    // hcat = relu([x@W0+b0 | agg1@W1+b1 | agg2@W2+b2])   (all dims multiples of 16)
    dim3 g1(gx, HH / 64);
    gemm_wmma_kernel<true><<<g1, 128, 0, stream>>>(x,    DD, W0, b0, hcat, 3 * HH, 0,      DD, HH, ntiles);
    gemm_wmma_kernel<true><<<g1, 128, 0, stream>>>(agg1, DD, W1, b1, hcat, 3 * HH, HH,     DD, HH, ntiles);
    gemm_wmma_kernel<true><<<g1, 128, 0, stream>>>(agg2, DD, W2, b2, hcat, 3 * HH, 2 * HH, DD, HH, ntiles);

    // out = hcat @ Wc + bc
    dim3 g2(gx, CC / 64);
    gemm_wmma_kernel<false><<<g2, 128, 0, stream>>>(hcat, 3 * HH, Wc, bc, out, CC, 0, 3 * HH, CC, ntiles);
}